// SSIM_79250736546681
// MI455X (gfx1250) — compile-verified
//
#include <hip/hip_runtime.h>

typedef __attribute__((ext_vector_type(2))) float v2f;
typedef __attribute__((ext_vector_type(8))) float v8f;

#define HTOT 512
#define WTOT 512
#define OUTD 502   // 512 - 11 + 1
#define TSTRIDE 36 // T row stride (floats): 16B-aligned rows, conflict-free banks

__device__ __forceinline__ unsigned fkey(float f) {
    unsigned b = __float_as_uint(f);
    return (b & 0x80000000u) ? ~b : (b | 0x80000000u);
}
__device__ __forceinline__ float fdecode(unsigned k) {
    unsigned b = (k & 0x80000000u) ? (k & 0x7FFFFFFFu) : ~k;
    return __uint_as_float(b);
}

__global__ void k_init(float* wsf, unsigned* wsu) {
    wsf[0] = 0.0f;
    wsu[1] = 0u;            // max key (very negative)
    wsu[2] = 0xFFFFFFFFu;   // min key (very positive)
}

__global__ void k_minmax(const float* __restrict__ x, int n, unsigned* wsu) {
    float mx = -3.4e38f, mn = 3.4e38f;
    for (int i = blockIdx.x * blockDim.x + threadIdx.x; i < n;
         i += gridDim.x * blockDim.x) {
        float v = x[i];
        mx = fmaxf(mx, v);
        mn = fminf(mn, v);
    }
    for (int off = 16; off; off >>= 1) {
        mx = fmaxf(mx, __shfl_down(mx, off, 32));
        mn = fminf(mn, __shfl_down(mn, off, 32));
    }
    __shared__ float smx[8], smn[8];
    int wid = threadIdx.x >> 5, lid = threadIdx.x & 31;
    if (lid == 0) { smx[wid] = mx; smn[wid] = mn; }
    __syncthreads();
    if (threadIdx.x == 0) {
        int nw = blockDim.x >> 5;
        for (int w = 1; w < nw; ++w) { mx = fmaxf(mx, smx[w]); mn = fminf(mn, smn[w]); }
        atomicMax(&wsu[1], fkey(mx));
        atomicMin(&wsu[2], fkey(mn));
    }
}

// One wave32 per 16x16 output tile. Separable 11-tap Gaussian conv done as
// banded matmuls on V_WMMA_F32_16X16X4_F32. All LDS traffic is unconditional
// (padded/clamped buffers) so no exec-mask churn in the hot loops.
__launch_bounds__(32)
__global__ void k_ssim(const float* __restrict__ img1, const float* __restrict__ img2,
                       const float* __restrict__ win, const unsigned* __restrict__ wsu,
                       float* wsf) {
    __shared__ float L1[32 * 28];          // rows 26..31 / cols 26..27 = clamped pad
    __shared__ float L2[32 * 28];
    __shared__ float T[5][16 * TSTRIDE];   // column-major: T[p][col*36 + row], rows 26+ zeroed
    __shared__ float Gs[44];               // Gs[16+k] = g[k] (k=0..10), zero elsewhere

    const int l = threadIdx.x;
    const int lm = l & 15;
    const int half = l >> 4;

    const int bidx  = blockIdx.x;
    const int plane = bidx >> 10;          // 32x32 tiles per (B,C) plane
    const int t     = bidx & 1023;
    const int ty = t >> 5, tx = t & 31;
    const int y0 = ty * 16, x0 = tx * 16;

    const float* p1 = img1 + (size_t)plane * HTOT * WTOT;
    const float* p2 = img2 + (size_t)plane * HTOT * WTOT;

    // Stage 26x26 region into 32x28 padded tiles (border-clamped, always finite).
    for (int idx = l; idx < 32 * 28; idx += 32) {
        int r = idx / 28, c = idx - r * 28;
        int gy = min(y0 + min(r, 25), HTOT - 1);
        int gx = min(x0 + min(c, 25), WTOT - 1);
        size_t o = (size_t)gy * WTOT + gx;
        L1[idx] = p1[o];
        L2[idx] = p2[o];
    }
    // Padded 1D gaussian: g[k] = w2[5][k]/sqrt(w2[5][5]); zero outside [0,10].
    for (int idx = l; idx < 44; idx += 32) {
        int k = idx - 16;
        Gs[idx] = ((unsigned)k <= 10u) ? win[55 + k] * rsqrtf(win[60]) : 0.0f;
    }
    __syncthreads();

    // Banded weight fragment, element (i=lm, j=4kc+2*half+v) -> g[j-i].
    // Shared by H-pass B-operand and V-pass A-operand. Unconditional b64 reads.
    v2f wf[7];
#pragma unroll
    for (int kc = 0; kc < 7; ++kc) {
        int d = 4 * kc + 2 * half - lm + 16;   // in [1, 42]
        wf[kc].x = Gs[d];
        wf[kc].y = Gs[d + 1];
    }

    // -------- Horizontal pass: tmp(26x16) = In(26x28) x Wh(28x16), 2 M-blocks --------
#pragma unroll
    for (int mb = 0; mb < 2; ++mb) {
        v8f a0 = {}, a1 = {}, a2 = {}, a3 = {}, a4 = {};
        const int base = (16 * mb + lm) * 28;
#pragma unroll
        for (int kc = 0; kc < 7; ++kc) {
            const int c0 = 4 * kc + 2 * half;
            float x1a = L1[base + c0], x1b = L1[base + c0 + 1];
            float x2a = L2[base + c0], x2b = L2[base + c0 + 1];
            v2f f1 = {x1a, x1b};
            v2f f2 = {x2a, x2b};
            v2f f3 = {x1a * x1a, x1b * x1b};
            v2f f4 = {x2a * x2a, x2b * x2b};
            v2f f5 = {x1a * x2a, x1b * x2b};
            v2f fb = wf[kc];
            a0 = __builtin_amdgcn_wmma_f32_16x16x4_f32(false, f1, false, fb, (short)0, a0, false, false);
            a1 = __builtin_amdgcn_wmma_f32_16x16x4_f32(false, f2, false, fb, (short)0, a1, false, false);
            a2 = __builtin_amdgcn_wmma_f32_16x16x4_f32(false, f3, false, fb, (short)0, a2, false, false);
            a3 = __builtin_amdgcn_wmma_f32_16x16x4_f32(false, f4, false, fb, (short)0, a3, false, false);
            a4 = __builtin_amdgcn_wmma_f32_16x16x4_f32(false, f5, false, fb, (short)0, a4, false, false);
        }
        // Column-major store: 8 consecutive floats per lane per plane (mergeable),
        // branch-free zeroing of rows >= 26.
        const int rbase = 16 * mb + 8 * half;
        const int cbase = lm * TSTRIDE + rbase;
#pragma unroll
        for (int v = 0; v < 8; ++v) {
            bool ok = (rbase + v) < 26;
            T[0][cbase + v] = ok ? a0[v] : 0.0f;
            T[1][cbase + v] = ok ? a1[v] : 0.0f;
            T[2][cbase + v] = ok ? a2[v] : 0.0f;
            T[3][cbase + v] = ok ? a3[v] : 0.0f;
            T[4][cbase + v] = ok ? a4[v] : 0.0f;
        }
    }
    __syncthreads();

    // -------- Vertical pass: out(16x16) = Gv(16x28) x tmp(28x16) --------
    v8f m1 = {}, m2 = {}, s11 = {}, s22 = {}, s12 = {};
    const int cb = lm * TSTRIDE;
#pragma unroll
    for (int kc = 0; kc < 7; ++kc) {
        const int j0 = 4 * kc + 2 * half;      // <= 27, rows 26/27 are zeros
        v2f fa = wf[kc];
        v2f b0 = {T[0][cb + j0], T[0][cb + j0 + 1]};
        v2f b1 = {T[1][cb + j0], T[1][cb + j0 + 1]};
        v2f b2 = {T[2][cb + j0], T[2][cb + j0 + 1]};
        v2f b3 = {T[3][cb + j0], T[3][cb + j0 + 1]};
        v2f b4 = {T[4][cb + j0], T[4][cb + j0 + 1]};
        m1  = __builtin_amdgcn_wmma_f32_16x16x4_f32(false, fa, false, b0, (short)0, m1,  false, false);
        m2  = __builtin_amdgcn_wmma_f32_16x16x4_f32(false, fa, false, b1, (short)0, m2,  false, false);
        s11 = __builtin_amdgcn_wmma_f32_16x16x4_f32(false, fa, false, b2, (short)0, s11, false, false);
        s22 = __builtin_amdgcn_wmma_f32_16x16x4_f32(false, fa, false, b3, (short)0, s22, false, false);
        s12 = __builtin_amdgcn_wmma_f32_16x16x4_f32(false, fa, false, b4, (short)0, s12, false, false);
    }

    // Dynamic-range constants from min/max reduction
    float maxv = fdecode(wsu[1]);
    float minv = fdecode(wsu[2]);
    float max_val = (maxv > 128.0f) ? 255.0f : 1.0f;
    float min_val = (minv < -0.5f) ? -1.0f : 0.0f;
    float Lr = max_val - min_val;
    float c1 = 0.0001f * Lr * Lr;   // (0.01*L)^2
    float c2 = 0.0009f * Lr * Lr;   // (0.03*L)^2

    // Pointwise SSIM + masked partial sum (each lane holds 8 pixels)
    float lsum = 0.0f;
#pragma unroll
    for (int v = 0; v < 8; ++v) {
        int y = y0 + 8 * half + v;
        int x = x0 + lm;
        float mu1 = m1[v], mu2 = m2[v];
        float mu1s = mu1 * mu1, mu2s = mu2 * mu2, m12 = mu1 * mu2;
        float sg1 = s11[v] - mu1s;
        float sg2 = s22[v] - mu2s;
        float sgx = s12[v] - m12;
        float num = (2.0f * m12 + c1) * (2.0f * sgx + c2);
        float den = (mu1s + mu2s + c1) * (sg1 + sg2 + c2);
        float val = num / den;
        lsum += (y < OUTD && x < OUTD) ? val : 0.0f;
    }
    for (int off = 16; off; off >>= 1) lsum += __shfl_down(lsum, off, 32);
    if (l == 0) atomicAdd(&wsf[0], lsum);
}

__global__ void k_final(const float* wsf, float* out) {
    // mean over 16*3*502*502 = 12,096,192 elements
    out[0] = wsf[0] * (1.0f / 12096192.0f);
}

extern "C" void kernel_launch(void* const* d_in, const int* in_sizes, int n_in,
                              void* d_out, int out_size, void* d_ws, size_t ws_size,
                              hipStream_t stream) {
    (void)n_in; (void)out_size; (void)ws_size;
    const float* img1 = (const float*)d_in[0];
    const float* img2 = (const float*)d_in[1];
    const float* win  = (const float*)d_in[2];
    float*    out = (float*)d_out;
    float*    wsf = (float*)d_ws;
    unsigned* wsu = (unsigned*)d_ws;

    k_init<<<1, 1, 0, stream>>>(wsf, wsu);
    k_minmax<<<1024, 256, 0, stream>>>(img1, in_sizes[0], wsu);
    // 16*3 planes * (32x32) tiles of 16x16 outputs
    k_ssim<<<48 * 1024, 32, 0, stream>>>(img1, img2, win, wsu, wsf);
    k_final<<<1, 1, 0, stream>>>(wsf, out);
}